// ConvolutionalFilter_39307540693514
// MI455X (gfx1250) — compile-verified
//
#include <hip/hip_runtime.h>

// ---- problem constants (from reference) ----
#define BATCH 32
#define CIN   64
#define HIN   112
#define WIN   112
#define FOUT  128
#define PADP  1

// ---- tiling ----
#define NPIX     64                  // pixel tile width (GEMM N)
#define CB       16                  // input channels staged per chunk
#define KCH      (CB * 9)            // 144 k-values per chunk
#define KSTR     148                 // padded k-row stride (16B-aligned rows: 148*4=592)
#define ISTRIDE  66                  // 64 pixels + 2 halo cols
#define ICH_STR  (3 * ISTRIDE)       // per-channel stride in raw tile (198)

#define WLDS_FLOATS (FOUT * KSTR)    // 18944 : wlds[f][k]
#define BLDS_FLOATS (NPIX * KSTR)    // 9472  : blds[p][k] (im2col)
#define IRAW_FLOATS (CB * ICH_STR)   // 3168  : raw halo tile
#define SMEM_BYTES  ((WLDS_FLOATS + BLDS_FLOATS + IRAW_FLOATS) * 4)  // 126336 B

typedef float v2f __attribute__((ext_vector_type(2)));
typedef float v4f __attribute__((ext_vector_type(4)));
typedef float v8f __attribute__((ext_vector_type(8)));

static __device__ __forceinline__ v8f wmma_f32(v2f a, v2f b, v8f c) {
    return __builtin_amdgcn_wmma_f32_16x16x4_f32(
        /*neg_a=*/false, a, /*neg_b=*/false, b,
        /*c_mod=*/(short)0, c, /*reuse_a=*/false, /*reuse_b=*/false);
}

__global__ __launch_bounds__(256)
void conv3x3_wmma_f32(const float* __restrict__ x,
                      const float* __restrict__ w,
                      float* __restrict__ y) {
    extern __shared__ float smem[];
    float* wlds = smem;                              // [FOUT][KSTR] weights, k-contiguous
    float* blds = smem + WLDS_FLOATS;                // [NPIX][KSTR] im2col, k-contiguous
    float* iraw = smem + WLDS_FLOATS + BLDS_FLOATS;  // [CB][3][ISTRIDE] raw halo

    const int tid    = threadIdx.x;
    const int lane   = tid & 31;
    const int wv     = tid >> 5;        // wave id 0..7
    const int laneLo = lane & 15;
    const int hi     = lane >> 4;       // half-wave: K pair +0/+1 vs +2/+3

    // block -> (pixel tile, output row, batch)
    int bid = blockIdx.x;
    const int pw  = bid & 1;
    bid >>= 1;
    const int oh  = bid % HIN;
    const int nb  = bid / HIN;
    const int ow0 = pw * NPIX;

    // wave tile: 4 waves across filters (M), 2 across pixels (N)
    const int fbase = (wv & 3) * 32;
    const int pbase = (wv >> 2) * 32;

    v8f acc[2][2] = {};

    // per-lane LDS fragment base pointers (constant across chunks)
    const float* aptr = &wlds[(fbase + laneLo) * KSTR + 2 * hi];
    const float* bptr = &blds[(pbase + laneLo) * KSTR + 2 * hi];

    // ---- staging ownership (all affine, no divides) ----
    const int wf    = tid >> 1;          // filter owned for weight staging
    const int wkb   = (tid & 1) * 72;    // k half
    const int sc    = tid >> 4;          // channel owned for halo staging
    const int scol  = tid & 15;          // column lane
    const int eg    = tid >> 6;          // channel group for im2col (0..3)
    const int ep    = tid & 63;          // pixel owned for im2col

    for (int c0 = 0; c0 < CIN; c0 += CB) {
        // ---- stage weights k-contiguous via b128: wlds[f*KSTR + k] ----
        {
            const float* gsrc = w + wf * (CIN * 9) + c0 * 9 + wkb;
            float*       ldst = wlds + wf * KSTR + wkb;
            #pragma unroll
            for (int i = 0; i < 18; ++i)
                *(v4f*)(ldst + i * 4) = *(const v4f*)(gsrc + i * 4);
        }
        // ---- stage raw input halo tile (zero-padded), affine indexing ----
        {
            const float* gx = x + ((nb * CIN + c0 + sc) * HIN) * WIN + (ow0 - PADP);
            float*       ld = iraw + sc * ICH_STR;
            #pragma unroll
            for (int kh = 0; kh < 3; ++kh) {
                int grow = oh - PADP + kh;
                bool rowok = (unsigned)grow < (unsigned)HIN;
                const float* grp = gx + grow * WIN;
                #pragma unroll
                for (int j = 0; j < 5; ++j) {
                    int col = scol + j * 16;
                    if (col < ISTRIDE) {
                        int gcol = ow0 - PADP + col;
                        float v = 0.0f;
                        if (rowok && (unsigned)gcol < (unsigned)WIN) v = grp[col];
                        ld[kh * ISTRIDE + col] = v;
                    }
                }
            }
        }
        __syncthreads();

        // ---- expand raw tile -> im2col, all-immediate offsets ----
        {
            const float* rsrc = iraw + ep + eg * (4 * ICH_STR);  // channel base g*4
            float*       bdst = blds + ep * KSTR + eg * (4 * 9);
            #pragma unroll
            for (int cc = 0; cc < 4; ++cc)
                #pragma unroll
                for (int kh = 0; kh < 3; ++kh)
                    #pragma unroll
                    for (int kw = 0; kw < 3; ++kw)
                        bdst[cc * 9 + kh * 3 + kw] =
                            rsrc[cc * ICH_STR + kh * ISTRIDE + kw];
        }
        __syncthreads();

        // prefetch next chunk's weights into cache while we compute
        if (c0 + CB < CIN)
            __builtin_prefetch(&w[wf * (CIN * 9) + (c0 + CB) * 9 + wkb], 0, 1);

        // ---- GEMM: 36 K-steps, pure ds_load_b64 + wmma ----
        #pragma unroll
        for (int kk = 0; kk < KCH; kk += 4) {
            v2f a0 = *(const v2f*)(aptr + kk);
            v2f a1 = *(const v2f*)(aptr + 16 * KSTR + kk);
            v2f b0 = *(const v2f*)(bptr + kk);
            v2f b1 = *(const v2f*)(bptr + 16 * KSTR + kk);
            acc[0][0] = wmma_f32(a0, b0, acc[0][0]);
            acc[0][1] = wmma_f32(a0, b1, acc[0][1]);
            acc[1][0] = wmma_f32(a1, b0, acc[1][0]);
            acc[1][1] = wmma_f32(a1, b1, acc[1][1]);
        }
        __syncthreads();
    }

    // ---- store: lane = pixel col (coalesced 16-lane runs), vgpr r = filter row ----
    #pragma unroll
    for (int n = 0; n < 2; ++n) {
        int pix = pbase + n * 16 + laneLo;
        int ow  = ow0 + pix;
        if (ow < WIN) {
            #pragma unroll
            for (int m = 0; m < 2; ++m) {
                #pragma unroll
                for (int r = 0; r < 8; ++r) {
                    int f = fbase + m * 16 + 8 * hi + r;
                    y[((nb * FOUT + f) * HIN + oh) * WIN + ow] = acc[m][n][r];
                }
            }
        }
    }
}

extern "C" void kernel_launch(void* const* d_in, const int* in_sizes, int n_in,
                              void* d_out, int out_size, void* d_ws, size_t ws_size,
                              hipStream_t stream) {
    (void)in_sizes; (void)n_in; (void)out_size; (void)d_ws; (void)ws_size;
    const float* x = (const float*)d_in[0];   // [32,64,112,112] fp32
    const float* w = (const float*)d_in[1];   // [128,64,3,3]    fp32
    float*       y = (float*)d_out;           // [32,128,112,112] fp32

    const int blocks = 2 /*W tiles*/ * HIN * BATCH;   // 7168
    conv3x3_wmma_f32<<<blocks, 256, SMEM_BYTES, stream>>>(x, w, y);
}